// BinaryDiceLoss_67542655697315
// MI455X (gfx1250) — compile-verified
//
#include <hip/hip_runtime.h>

// BCE-loss grand reduction for MI455X (gfx1250, wave32).
// Bandwidth-bound: 256 MiB read-once -> NT b128 streaming loads; WMMA
// f32_16x16x4 used as the in-wave reduction primitive; deterministic
// two-pass (block partials in d_ws, single-block finish).

typedef __attribute__((ext_vector_type(4))) float v4f;
typedef __attribute__((ext_vector_type(2))) float v2f;
typedef __attribute__((ext_vector_type(8))) float v8f;

#define LOG_CLAMP (-100.0f)

__device__ __forceinline__ float bce_inner(float p, float t) {
    // t*log(p) + (1-t)*log(1-p), each log clamped at -100 (torch BCELoss)
    float lp = fmaxf(__logf(p), LOG_CLAMP);
    float l1 = fmaxf(__logf(1.0f - p), LOG_CLAMP);
    return fmaf(t, lp - l1, l1);
}

// Full-wave (32 lane) sum using V_WMMA_F32_16X16X4_F32.
// A layout (16x4 f32): lanes 0-15 -> VGPR0=A[m][0], VGPR1=A[m][1];
//                      lanes 16-31 -> VGPR0=A[m][2], VGPR1=A[m][3].
// With a = {v, 0} and B = ones(4x16): D[m][n] = v[m] + v[m+16].
// In-lane sum of the 8 D VGPRs gives the two half-wave sums; one
// shfl_xor(16) combines them. EXEC must be all 1s at the call site.
__device__ __forceinline__ float wave_reduce_wmma(float v) {
    v2f a; a.x = v;    a.y = 0.0f;
    v2f b; b.x = 1.0f; b.y = 1.0f;
    v8f c = {};
    v8f d = __builtin_amdgcn_wmma_f32_16x16x4_f32(
        /*neg_a=*/false, a, /*neg_b=*/false, b,
        /*c_mod=*/(short)0, c, /*reuse_a=*/false, /*reuse_b=*/false);
    float s = ((d[0] + d[1]) + (d[2] + d[3])) + ((d[4] + d[5]) + (d[6] + d[7]));
    s += __shfl_xor(s, 16, 32);
    return s;
}

__device__ __forceinline__ float block_reduce(float acc, int tid) {
    float ws = wave_reduce_wmma(acc);
    __shared__ float smem[8];
    int lane = tid & 31;
    int wave = tid >> 5;
    if (lane == 0) smem[wave] = ws;
    __syncthreads();
    float s = 0.0f;
    if (tid == 0) {
        #pragma unroll
        for (int w = 0; w < 8; ++w) s += smem[w];
    }
    return s;  // valid in thread 0 only
}

__global__ void __launch_bounds__(256)
bce_partial_kernel(const float* __restrict__ predict,
                   const float* __restrict__ target,
                   float* __restrict__ partial,
                   long long n4, long long n) {
    long long tid    = (long long)blockIdx.x * blockDim.x + threadIdx.x;
    long long stride = (long long)gridDim.x * blockDim.x;

    const v4f* p4 = (const v4f*)predict;
    const v4f* t4 = (const v4f*)target;

    float acc = 0.0f;
    for (long long i = tid; i < n4; i += stride) {
        v4f p = __builtin_nontemporal_load(p4 + i);   // global_load_b128 NT
        v4f t = __builtin_nontemporal_load(t4 + i);
        acc += bce_inner(p.x, t.x);
        acc += bce_inner(p.y, t.y);
        acc += bce_inner(p.z, t.z);
        acc += bce_inner(p.w, t.w);
    }
    // scalar tail (none for this shape, but keep it correct)
    if (tid == 0) {
        for (long long i = n4 * 4; i < n; ++i)
            acc += bce_inner(predict[i], target[i]);
    }

    float s = block_reduce(acc, threadIdx.x);
    if (threadIdx.x == 0) partial[blockIdx.x] = s;
}

__global__ void __launch_bounds__(256)
bce_final_kernel(const float* __restrict__ partial,
                 float* __restrict__ out, int nblocks) {
    float acc = 0.0f;
    for (int i = threadIdx.x; i < nblocks; i += blockDim.x)
        acc += partial[i];
    float s = block_reduce(acc, threadIdx.x);
    if (threadIdx.x == 0) out[0] = -s;  // loss = -(sum of inner terms)
}

extern "C" void kernel_launch(void* const* d_in, const int* in_sizes, int n_in,
                              void* d_out, int out_size, void* d_ws, size_t ws_size,
                              hipStream_t stream) {
    const float* predict = (const float*)d_in[0];
    const float* target  = (const float*)d_in[1];
    float* out           = (float*)d_out;
    float* partial       = (float*)d_ws;

    long long n  = (long long)in_sizes[0];
    long long n4 = n >> 2;

    const int threads = 256;  // 8 wave32s per block
    long long want = (n4 + threads - 1) / threads;
    int blocks = (int)(want < 4096 ? want : 4096);
    size_t ws_cap = ws_size / sizeof(float);
    if ((size_t)blocks > ws_cap) blocks = (int)ws_cap;
    if (blocks < 1) blocks = 1;

    bce_partial_kernel<<<blocks, threads, 0, stream>>>(predict, target, partial, n4, n);
    bce_final_kernel<<<1, threads, 0, stream>>>(partial, out, blocks);
}